// Decoder_45715631899300
// MI455X (gfx1250) — compile-verified
//
#include <hip/hip_runtime.h>
#include <stdint.h>

#define B_      8
#define N_      1000
#define C_      81
#define NP2     1024
#define MAXT    200
#define CAP     (C_ * MAXT)     /* 16200 max kept per batch */
#define SCORE_T 0.5f
#define IOU_T   0.5f

typedef unsigned int v4u __attribute__((ext_vector_type(4)));
typedef int          v8i __attribute__((ext_vector_type(8)));
typedef int          v4i __attribute__((ext_vector_type(4)));

// ---------------------------------------------------------------------------
// Tensor Data Mover: DMA a contiguous run of 4-byte elements (1-row tile)
// from global memory into LDS. Descriptor packing per CDNA5 ISA §8.
// ---------------------------------------------------------------------------
__device__ __forceinline__ void tdm_load_to_lds(const void* gsrc, void* ldsDst,
                                                unsigned nElem4B) {
  unsigned long long ga = (unsigned long long)(uintptr_t)gsrc;
  unsigned la = (unsigned)(uintptr_t)ldsDst;   // LDS aperture: low 32 bits
  v4u g0;
  g0.x = 1u;                                          // count=1, is_restore=0
  g0.y = la;                                          // lds_addr
  g0.z = (unsigned)(ga & 0xffffffffull);              // global_addr[31:0]
  g0.w = (unsigned)((ga >> 32) & 0x01ffffffull)       // global_addr[56:32]
       | (2u << 30);                                  // type=2 ("image")
  v8i g1;
  g1[0] = 0x00020000;                                 // data_size=2 (4 bytes)
  g1[1] = (int)((nElem4B & 0xffffu) << 16);           // tensor_dim0[15:0]
  g1[2] = (int)((nElem4B >> 16) | (1u << 16));        // tensor_dim0[31:16] | tensor_dim1=1
  g1[3] = (int)((nElem4B & 0xffffu) << 16);           // tile_dim0
  g1[4] = 1;                                          // tile_dim1=1, tile_dim2=0
  g1[5] = (int)nElem4B;                               // tensor_dim0_stride[31:0]
  g1[6] = 0;
  g1[7] = 0;
  v4i gz4 = {0, 0, 0, 0};
#if defined(__clang_major__) && (__clang_major__ >= 23)
  v8i gz8 = {0, 0, 0, 0, 0, 0, 0, 0};
  __builtin_amdgcn_tensor_load_to_lds(g0, g1, gz4, gz4, gz8, 0);
#else
  __builtin_amdgcn_tensor_load_to_lds(g0, g1, gz4, gz4, 0);
#endif
}

// ---------------------------------------------------------------------------
// Kernel 1: background flags (argmax over classes == 0) + zero batch counters
// ---------------------------------------------------------------------------
__global__ void prep_kernel(const float* __restrict__ probs,
                            int* __restrict__ bg, int* __restrict__ counters) {
  int t = blockIdx.x * blockDim.x + threadIdx.x;
  if (t < B_) counters[t] = 0;
  if (t >= B_ * N_) return;
  const float* p = probs + (size_t)t * C_;
  float m = p[0];
  int am = 0;
  for (int c = 1; c < C_; ++c) {
    float v = p[c];
    if (v > m) { m = v; am = c; }     // first-max wins ties (jnp.argmax)
  }
  bg[t] = (am == 0) ? 1 : 0;
}

// ---------------------------------------------------------------------------
// Kernel 2: decode boxes -> [B,C,N,4], masked scores -> [B,C,N]
// ---------------------------------------------------------------------------
__global__ void decode_kernel(const float* __restrict__ rois,
                              const float* __restrict__ deltas,
                              const float* __restrict__ probs,
                              const int* __restrict__ bg,
                              float* __restrict__ boxes,
                              float* __restrict__ scores) {
  int t = blockIdx.x * blockDim.x + threadIdx.x;
  if (t >= B_ * N_ * C_) return;
  int c = t % C_;
  int bn = t / C_;
  int n = bn % N_;
  int b = bn / N_;
  const float* r = rois + (size_t)bn * 4;
  float y1 = r[0], x1 = r[1], y2 = r[2], x2 = r[3];
  float h = y2 - y1, w = x2 - x1;
  float cy = y1 + 0.5f * h, cx = x1 + 0.5f * w;
  const float* d = deltas + ((size_t)bn * C_ + c) * 4;
  float ncy = d[0] * 0.1f * h + cy;          // VARIANCES = 0.1,0.1,0.2,0.2
  float ncx = d[1] * 0.1f * w + cx;
  float nh = expf(d[2] * 0.2f) * h;
  float nw = expf(d[3] * 0.2f) * w;
  size_t o = ((size_t)b * C_ + c) * (size_t)N_ + (size_t)n;
  float* ob = boxes + o * 4;
  ob[0] = ncy - 0.5f * nh;
  ob[1] = ncx - 0.5f * nw;
  ob[2] = ncy + 0.5f * nh;
  ob[3] = ncx + 0.5f * nw;
  scores[o] = bg[bn] ? 0.0f : probs[(size_t)bn * C_ + c];
}

// ---------------------------------------------------------------------------
// Kernel 3: per-(b,c) lane: TDM-stage boxes to LDS, bitonic sort keys,
// greedy NMS, 200-per-class cap, append kept entries to per-batch list.
// ---------------------------------------------------------------------------
__global__ __launch_bounds__(512)
void nms_kernel(const float* __restrict__ boxes,
                const float* __restrict__ scores,
                float* __restrict__ lists, int* __restrict__ counters) {
  __shared__ float s_sc[NP2];
  __shared__ int   s_id[NP2];
  __shared__ float s_box[NP2 * 4];       // TDM destination (original order)
  __shared__ int   s_keep[NP2];
  __shared__ int   s_scan[NP2];
  __shared__ int   s_base;

  const int lane = blockIdx.x;
  const int b = lane / C_;
  const int c = lane % C_;
  const int tid = threadIdx.x;
  const float* gbox = boxes + (size_t)lane * N_ * 4;
  const float* gsc  = scores + (size_t)lane * N_;

  // Start the DMA; the wave sorts score keys while boxes stream into LDS.
  if (tid == 0) tdm_load_to_lds(gbox, (void*)s_box, N_ * 4);

  for (int t = tid; t < NP2; t += 512) {
    s_sc[t] = (t < N_) ? gsc[t] : -1.0e30f;
    s_id[t] = t;
  }
  __syncthreads();

  // Bitonic sort, descending by score, ascending index on ties (stable-like).
  for (int k = 2; k <= NP2; k <<= 1) {
    for (int j = k >> 1; j > 0; j >>= 1) {
      for (int t = tid; t < NP2; t += 512) {
        int ixj = t ^ j;
        if (ixj > t) {
          float sA = s_sc[t], sB = s_sc[ixj];
          int iA = s_id[t], iB = s_id[ixj];
          bool aFirst = (sA > sB) || ((sA == sB) && (iA < iB));
          bool descBlk = ((t & k) == 0);
          if (descBlk ? !aFirst : aFirst) {
            s_sc[t] = sB; s_sc[ixj] = sA;
            s_id[t] = iB; s_id[ixj] = iA;
          }
        }
      }
      __syncthreads();
    }
  }

  // Boxes must be resident in LDS before first use.
  __builtin_amdgcn_s_wait_tensorcnt(0);
  __syncthreads();

  const int t0 = tid, t1 = tid + 512;
  // Register-cache this thread's two candidates (sorted order).
  int bi0 = s_id[t0] * 4, bi1 = s_id[t1] * 4;
  float c0y1 = s_box[bi0 + 0], c0x1 = s_box[bi0 + 1];
  float c0y2 = s_box[bi0 + 2], c0x2 = s_box[bi0 + 3];
  float c1y1 = s_box[bi1 + 0], c1x1 = s_box[bi1 + 1];
  float c1y2 = s_box[bi1 + 2], c1x2 = s_box[bi1 + 3];
  float ar0 = (c0y2 - c0y1) * (c0x2 - c0x1);
  float ar1 = (c1y2 - c1y1) * (c1x2 - c1x1);

  s_keep[t0] = (s_sc[t0] > SCORE_T) ? 1 : 0;   // score filtering before NMS
  s_keep[t1] = (s_sc[t1] > SCORE_T) ? 1 : 0;
  __syncthreads();

  // Greedy NMS (uniform control flow; early exit once sorted scores <= 0.5).
  for (int i = 0; i < NP2; ++i) {
    if (s_sc[i] <= SCORE_T) break;             // uniform across the block
    if (s_keep[i]) {
      int bi = s_id[i] * 4;
      float iy1 = s_box[bi + 0], ix1 = s_box[bi + 1];
      float iy2 = s_box[bi + 2], ix2 = s_box[bi + 3];
      float ia = (iy2 - iy1) * (ix2 - ix1);
      if (t0 > i && s_keep[t0]) {
        float yy1 = fmaxf(iy1, c0y1), xx1 = fmaxf(ix1, c0x1);
        float yy2 = fminf(iy2, c0y2), xx2 = fminf(ix2, c0x2);
        float inter = fmaxf(yy2 - yy1, 0.0f) * fmaxf(xx2 - xx1, 0.0f);
        float iou = inter / (ia + ar0 - inter + 1e-8f);
        if (iou > IOU_T) s_keep[t0] = 0;
      }
      if (t1 > i && s_keep[t1]) {
        float yy1 = fmaxf(iy1, c1y1), xx1 = fmaxf(ix1, c1x1);
        float yy2 = fminf(iy2, c1y2), xx2 = fminf(ix2, c1x2);
        float inter = fmaxf(yy2 - yy1, 0.0f) * fmaxf(xx2 - xx1, 0.0f);
        float iou = inter / (ia + ar1 - inter + 1e-8f);
        if (iou > IOU_T) s_keep[t1] = 0;
      }
    }
    __syncthreads();
  }

  // Inclusive scan of keep flags (Hillis-Steele) for the per-class 200 cap.
  s_scan[t0] = s_keep[t0];
  s_scan[t1] = s_keep[t1];
  __syncthreads();
  for (int off = 1; off < NP2; off <<= 1) {
    int v0 = (t0 >= off) ? s_scan[t0 - off] : 0;
    int v1 = (t1 >= off) ? s_scan[t1 - off] : 0;
    __syncthreads();
    s_scan[t0] += v0;
    s_scan[t1] += v1;
    __syncthreads();
  }

  int total = s_scan[NP2 - 1];
  int nEmit = total < MAXT ? total : MAXT;
  if (tid == 0) s_base = atomicAdd(&counters[b], nEmit);
  __syncthreads();

#pragma unroll
  for (int q = 0; q < 2; ++q) {
    int t = (q == 0) ? t0 : t1;
    if (s_keep[t] && s_scan[t] <= MAXT) {
      int pos = s_base + s_scan[t] - 1;
      float* e = lists + ((size_t)b * CAP + (size_t)pos) * 8;
      int bi = s_id[t] * 4;
      e[0] = s_box[bi + 0];
      e[1] = s_box[bi + 1];
      e[2] = s_box[bi + 2];
      e[3] = s_box[bi + 3];
      e[4] = s_sc[t];
      e[5] = __int_as_float(c * N_ + t);   // flat key == reference flat index
    }
  }
}

// ---------------------------------------------------------------------------
// Kernel 4: per-batch combined top-200 over the kept list; write outputs.
// ---------------------------------------------------------------------------
__global__ __launch_bounds__(256)
void topk_kernel(const float* __restrict__ lists,
                 const int* __restrict__ counters, float* __restrict__ out) {
  __shared__ float rs[256];
  __shared__ int   rk[256];
  __shared__ int   ri[256];
  __shared__ unsigned taken[(CAP + 31) / 32];

  int b = blockIdx.x, tid = threadIdx.x;
  const float* list = lists + (size_t)b * CAP * 8;
  int K = counters[b];
  if (K > CAP) K = CAP;
  if (K < 0) K = 0;
  for (int i = tid; i < (CAP + 31) / 32; i += 256) taken[i] = 0u;
  __syncthreads();

  float* ob = out + (size_t)b * MAXT * 4;
  float* oc = out + (size_t)B_ * MAXT * 4 + (size_t)b * MAXT;
  float* os = out + (size_t)B_ * MAXT * 5 + (size_t)b * MAXT;

  for (int sel = 0; sel < MAXT; ++sel) {
    float bs = 0.0f;
    int bk = 0x7fffffff;
    int bi = -1;
    for (int i = tid; i < K; i += 256) {
      if (taken[i >> 5] & (1u << (i & 31))) continue;
      float s = list[(size_t)i * 8 + 4];
      if (s <= 0.0f) continue;
      int k = __float_as_int(list[(size_t)i * 8 + 5]);
      if (bi < 0 || s > bs || (s == bs && k < bk)) { bs = s; bk = k; bi = i; }
    }
    rs[tid] = bs; rk[tid] = bk; ri[tid] = bi;
    __syncthreads();
    for (int off = 128; off > 0; off >>= 1) {
      if (tid < off) {
        float s2 = rs[tid + off];
        int k2 = rk[tid + off], i2 = ri[tid + off];
        bool better = (i2 >= 0) &&
                      (ri[tid] < 0 || s2 > rs[tid] ||
                       (s2 == rs[tid] && k2 < rk[tid]));
        if (better) { rs[tid] = s2; rk[tid] = k2; ri[tid] = i2; }
      }
      __syncthreads();
    }
    if (tid == 0) {
      int i = ri[0];
      if (i >= 0) {
        const float* e = list + (size_t)i * 8;
        ob[sel * 4 + 0] = fminf(fmaxf(e[0], 0.0f), 1.0f);
        ob[sel * 4 + 1] = fminf(fmaxf(e[1], 0.0f), 1.0f);
        ob[sel * 4 + 2] = fminf(fmaxf(e[2], 0.0f), 1.0f);
        ob[sel * 4 + 3] = fminf(fmaxf(e[3], 0.0f), 1.0f);
        oc[sel] = (float)(rk[0] / N_);
        os[sel] = rs[0];
        taken[i >> 5] |= (1u << (i & 31));
      } else {
        ob[sel * 4 + 0] = 0.0f; ob[sel * 4 + 1] = 0.0f;
        ob[sel * 4 + 2] = 0.0f; ob[sel * 4 + 3] = 0.0f;
        oc[sel] = 0.0f; os[sel] = 0.0f;
      }
    }
    __syncthreads();
  }
}

// ---------------------------------------------------------------------------
extern "C" void kernel_launch(void* const* d_in, const int* in_sizes, int n_in,
                              void* d_out, int out_size, void* d_ws,
                              size_t ws_size, hipStream_t stream) {
  (void)in_sizes; (void)n_in; (void)out_size; (void)ws_size;
  const float* rois   = (const float*)d_in[0];
  const float* deltas = (const float*)d_in[1];
  const float* probs  = (const float*)d_in[2];
  float* out = (float*)d_out;

  char* ws = (char*)d_ws;
  size_t off = 0;
  float* boxes  = (float*)(ws + off); off += (size_t)B_ * C_ * N_ * 4 * sizeof(float);
  float* scores = (float*)(ws + off); off += (size_t)B_ * C_ * N_ * sizeof(float);
  int*   bg     = (int*)(ws + off);   off += (size_t)B_ * N_ * sizeof(int);
  int*   ctrs   = (int*)(ws + off);   off += 256;
  float* lists  = (float*)(ws + off); // B * CAP * 8 floats (~4.15 MB)

  prep_kernel<<<(B_ * N_ + 255) / 256, 256, 0, stream>>>(probs, bg, ctrs);
  decode_kernel<<<(B_ * N_ * C_ + 255) / 256, 256, 0, stream>>>(
      rois, deltas, probs, bg, boxes, scores);
  nms_kernel<<<B_ * C_, 512, 0, stream>>>(boxes, scores, lists, ctrs);
  topk_kernel<<<B_, 256, 0, stream>>>(lists, ctrs, out);
}